// Conv2dAMPS_22333829939474
// MI455X (gfx1250) — compile-verified
//
#include <hip/hip_runtime.h>

typedef float v2f __attribute__((ext_vector_type(2)));
typedef float v8f __attribute__((ext_vector_type(8)));
typedef unsigned int u32x4 __attribute__((ext_vector_type(4)));
typedef int i32x4 __attribute__((ext_vector_type(4)));
typedef int i32x8 __attribute__((ext_vector_type(8)));

#define Bn 16
#define Cn 64
#define Hn 32
#define Wn 32
#define On 64
#define OH 31
#define OW 31
#define NPATCH (Bn*OH*OW)        // 15376
#define NT32 ((NPATCH + 31)/32)  // 481 tiles of 32 patches
#define WAVES_PER_BLOCK 4
#define WROW 68                  // 64 + 4 pad floats (TDM pad) -> conflict-free ds_load_b64
#define VROW 65                  // padded v row -> conflict-free broadcast reads

// ---- Tensor Data Mover: DMA one contiguous 16KB W-slice (tensors[w,k,:,:])
// into LDS as 64 rows x 68 floats (4-DWORD pad every 64 DWORDs).
__device__ __forceinline__ void tdm_load_slice(const float* gsrc, unsigned lds_byte_off)
{
    unsigned long long ga = (unsigned long long)(uintptr_t)gsrc;
    u32x4 g0;
    g0[0] = 1u;                                    // count=1, is_restore=0, gather_mode=0
    g0[1] = lds_byte_off;                          // D#.lds_addr
    g0[2] = (unsigned)(ga & 0xffffffffu);          // global_addr[31:0]
    g0[3] = (unsigned)((ga >> 32) & 0x01ffffffu)   // global_addr[56:32]
          | (2u << 30);                            // type = 2 ("image")
    i32x8 g1;
    g1[0] = (int)((2u << 16)                       // data_size = 4 bytes
                | (1u << 20)                       // pad_enable
                | (5u << 22)                       // pad_interval: 64 DWORDs
                | (3u << 25));                     // pad_amount: 4 DWORDs
    g1[1] = (int)(4096u << 16);                    // tensor_dim0 = 4096 (bits [79:48] low half)
    g1[2] = 0;
    g1[3] = (int)(4096u << 16);                    // tile_dim0 = 4096 (bits [127:112])
    g1[4] = 0;                                     // tile_dim1/2 = 0 (unused)
    g1[5] = 4096;                                  // tensor_dim0_stride (benign for 1-D tile)
    g1[6] = 0;
    g1[7] = 0;
    i32x4 gz = {0, 0, 0, 0};
#if defined(__clang_major__) && (__clang_major__ >= 23)
    i32x8 gz8 = {0, 0, 0, 0, 0, 0, 0, 0};
    __builtin_amdgcn_tensor_load_to_lds(g0, g1, gz, gz, gz8, 0);
#else
    __builtin_amdgcn_tensor_load_to_lds(g0, g1, gz, gz, 0);
#endif
}

// Row-vector chain per patch:  out_row = e0^T * M0 * M1 * M2 * M3,
// M_w = sum_p tensors[w,:,:,p] * E_w[n,p].
// Stage 0:  v = E0 (32x64) @ B0 (64x64),  B0[p,j] = tensors[0,0,j,p]
// Stage w:  v' = Z (32x4096) @ W'_w (4096x64), Z[n,(k,p)] = v[n,k]*E_w[n,p].
// M=32 per wave as two 16-row A blocks sharing every B fragment (2 WMMA / ds_load_b64).
__global__ __launch_bounds__(128) void conv_chain_wmma_f32(
    const float* __restrict__ in, const float* __restrict__ tn,
    float* __restrict__ out)
{
    __shared__ float wlds[2][64 * WROW];                    // double-buffered W slice
    __shared__ float vbuf_all[WAVES_PER_BLOCK][32 * VROW];  // per-wave v transpose staging

    const int lane = threadIdx.x & 31;
    const int wave = threadIdx.x >> 5;
    const int tile = blockIdx.x * WAVES_PER_BLOCK + wave;
    const bool active = (tile < NT32);          // wave-uniform; wave 0 always active

    const int m = lane & 15;                    // M row within a 16-row block
    const int h = lane >> 4;                    // lane-half: K sub-offset 2*h
    float* vb = vbuf_all[wave];

    // Patch coords for the two 16-row blocks (clamped for the ragged last tile)
    int ebase[2];
#pragma unroll
    for (int s = 0; s < 2; ++s) {
        int nn = (active ? tile : 0) * 32 + s * 16 + m;
        nn = nn < NPATCH ? nn : NPATCH - 1;
        const int pb = nn / (OH * OW);
        const int rr = nn % (OH * OW);
        ebase[s] = pb * (Cn * Hn * Wn) + (rr / OW) * Wn + (rr % OW);
    }

    // Per-lane float offset of this lane's B float2 inside the padded LDS slice
    int jbl[4];
#pragma unroll
    for (int jt = 0; jt < 4; ++jt) jbl[jt] = ((jt << 4) + m) * WROW + 2 * h;

    v8f acc[2][4];
#pragma unroll
    for (int s = 0; s < 2; ++s)
#pragma unroll
        for (int jt = 0; jt < 4; ++jt)
#pragma unroll
            for (int e = 0; e < 8; ++e) acc[s][jt][e] = 0.0f;

    v2f ea[2][16];   // E_w[row, 4t+2h .. +1] for both patch blocks

    // ---------------- stage 0: v = E0 @ B0 (K = 64, B TDM-staged) ----------------
    if (wave == 0) {                            // DMA tensors[0,0,:,:] -> wlds[0]
        tdm_load_slice(tn, (unsigned)(uintptr_t)&wlds[0][0]);
        __builtin_amdgcn_s_wait_tensorcnt(0);
    }
    if (active) {
#pragma unroll
        for (int s = 0; s < 2; ++s)
#pragma unroll
            for (int t = 0; t < 16; ++t) {
                const int p0 = 4 * t + 2 * h;
                v2f e;
                e.x = in[ebase[s] + p0 * (Hn * Wn)];
                e.y = in[ebase[s] + (p0 + 1) * (Hn * Wn)];
                ea[s][t] = e;
            }
    }
    __syncthreads();
    if (active) {
#pragma unroll
        for (int kb = 0; kb < 16; ++kb) {
            const v2f a0 = ea[0][kb], a1 = ea[1][kb];
#pragma unroll
            for (int jt = 0; jt < 4; ++jt) {
                v2f b = *(const v2f*)(&wlds[0][0] + jbl[jt] + 4 * kb);  // ds_load_b64
                acc[0][jt] = __builtin_amdgcn_wmma_f32_16x16x4_f32(
                    false, a0, false, b, (short)0, acc[0][jt], false, false);
                acc[1][jt] = __builtin_amdgcn_wmma_f32_16x16x4_f32(
                    false, a1, false, b, (short)0, acc[1][jt], false, false);
            }
        }
    }
    __syncthreads();   // all reads of wlds[0] done before w-loop prologue refills it

    // ---------------- stages 1..3: v = Z @ W'_w (K = 4096) ----------------
#pragma unroll 1
    for (int w = 1; w < 4; ++w) {
        const float* Wt = tn + w * (On * On * Cn);

        if (active) {
            // Spill v (D layout: lane l, vgpr r of acc[s][jt] = v[s*16+r+8h, jt*16+m])
#pragma unroll
            for (int s = 0; s < 2; ++s)
#pragma unroll
                for (int jt = 0; jt < 4; ++jt)
#pragma unroll
                    for (int r = 0; r < 8; ++r)
                        vb[(s * 16 + r + 8 * h) * VROW + (jt << 4) + m] = acc[s][jt][r];

            // Load E_w for this tap (di*Wn + dj offset from the base patch)
            const int di = w >> 1, dj = w & 1;
#pragma unroll
            for (int s = 0; s < 2; ++s)
#pragma unroll
                for (int t = 0; t < 16; ++t) {
                    const int p0 = 4 * t + 2 * h;
                    v2f e;
                    e.x = in[ebase[s] + di * Wn + dj + p0 * (Hn * Wn)];
                    e.y = in[ebase[s] + di * Wn + dj + (p0 + 1) * (Hn * Wn)];
                    ea[s][t] = e;
                }

#pragma unroll
            for (int s = 0; s < 2; ++s)
#pragma unroll
                for (int jt = 0; jt < 4; ++jt)
#pragma unroll
                    for (int e = 0; e < 8; ++e) acc[s][jt][e] = 0.0f;
        }

        // Pipeline prologue: DMA k=0 slice into buffer 0
        if (wave == 0) {
            tdm_load_slice(Wt, (unsigned)(uintptr_t)&wlds[0][0]);
            __builtin_amdgcn_s_wait_tensorcnt(0);
        }
        __syncthreads();

#pragma unroll 1
        for (int k = 0; k < 64; ++k) {
            const int cur = k & 1;
            // Overlap: DMA slice k+1 while computing on slice k (WAR safe via barriers)
            if (wave == 0 && k < 63)
                tdm_load_slice(Wt + (k + 1) * (On * Cn),
                               (unsigned)(uintptr_t)&wlds[cur ^ 1][0]);

            if (active) {
                const float* wb = &wlds[cur][0];
                const float vk0 = vb[m * VROW + k];          // v[m, k]
                const float vk1 = vb[(m + 16) * VROW + k];   // v[m+16, k]
#pragma unroll
                for (int kb = 0; kb < 16; ++kb) {
                    v2f a0, a1;
                    a0.x = vk0 * ea[0][kb].x;  a0.y = vk0 * ea[0][kb].y;
                    a1.x = vk1 * ea[1][kb].x;  a1.y = vk1 * ea[1][kb].y;
#pragma unroll
                    for (int jt = 0; jt < 4; ++jt) {
                        v2f b = *(const v2f*)(wb + jbl[jt] + 4 * kb);   // ds_load_b64
                        acc[0][jt] = __builtin_amdgcn_wmma_f32_16x16x4_f32(
                            false, a0, false, b, (short)0, acc[0][jt], false, false);
                        acc[1][jt] = __builtin_amdgcn_wmma_f32_16x16x4_f32(
                            false, a1, false, b, (short)0, acc[1][jt], false, false);
                    }
                }
            }

            if (wave == 0)
                __builtin_amdgcn_s_wait_tensorcnt(0);        // slice k+1 landed
            __syncthreads();                                 // all waves done with slice k
        }
    }

    if (!active) return;

    // ---------------- store: out[b, j, oi, oj] ----------------
#pragma unroll
    for (int s = 0; s < 2; ++s)
#pragma unroll
        for (int r = 0; r < 8; ++r) {
            const int nn = tile * 32 + s * 16 + 8 * h + r;
            if (nn < NPATCH) {
                const int b2 = nn / (OH * OW);
                const int rr = nn % (OH * OW);
                const int i2 = rr / OW;
                const int j2 = rr % OW;
                float* op = out + (b2 * On) * (OH * OW) + i2 * OW + j2;
#pragma unroll
                for (int jt = 0; jt < 4; ++jt) {
                    const int j = (jt << 4) + m;
                    op[j * (OH * OW)] = acc[s][jt][r];
                }
            }
        }
}

extern "C" void kernel_launch(void* const* d_in, const int* in_sizes, int n_in,
                              void* d_out, int out_size, void* d_ws, size_t ws_size,
                              hipStream_t stream) {
    const float* in = (const float*)d_in[0];   // (16,64,32,32) f32
    const float* tn = (const float*)d_in[1];   // (4,64,64,64)  f32
    float* out = (float*)d_out;                // (16,64,31,31) f32

    const int blocks = (NT32 + WAVES_PER_BLOCK - 1) / WAVES_PER_BLOCK;   // 121
    conv_chain_wmma_f32<<<blocks, 128, 0, stream>>>(in, tn, out);
}